// HPUBenchmarkModel_53446573031600
// MI455X (gfx1250) — compile-verified
//
#include <hip/hip_runtime.h>
#include <stdint.h>

#define NSEQ 64
#define NH   32
#define NKVH 8
#define GQA  4
#define HD   128
#define BS   16
#define MAXB 128   // MAX_SEQ_LEN / BLOCK_SIZE

typedef __attribute__((ext_vector_type(2))) float v2f;
typedef __attribute__((ext_vector_type(8))) float v8f;
typedef int v4i_g __attribute__((vector_size(16)));   // matches builtin param type

// ---- CDNA5 async global->LDS copy (ASYNCcnt-tracked, no VGPR staging) ----
// Signature (from hipcc diagnostic): (AS1 v4i*, AS3 v4i*, imm offset, imm cpol)
#if __has_builtin(__builtin_amdgcn_global_load_async_to_lds_b128)
#define ASYNC_B128(g, l, imm)                                                   \
    __builtin_amdgcn_global_load_async_to_lds_b128(                             \
        (__attribute__((address_space(1))) v4i_g*)(uintptr_t)(g),               \
        (__attribute__((address_space(3))) v4i_g*)(unsigned)(uintptr_t)(l),     \
        (imm), 0)
#else
#define ASYNC_B128(g, l, imm)                                                   \
    asm volatile("global_load_async_to_lds_b128 %0, %1, off offset:%2"          \
                 :: "v"((unsigned)(uintptr_t)(l)),                              \
                    "v"((const void*)(g)),                                      \
                    "n"(imm)                                                    \
                 : "memory")
#endif

// s_wait_asynccnt with a compiler memory barrier: keeps LDS reads below the
// wait and makes the async writes "visible" to the optimizer.
#define WAIT_ASYNC(n) asm volatile("s_wait_asynccnt %0" :: "n"(n) : "memory")

// Issue 16 b128 async loads covering one 8KB tile (per-lane base + imm offs;
// the single instruction offset applies to both global and LDS addresses).
#define ISSUE4(g, l, base)                                                      \
    ASYNC_B128((g), (l), (base));                                               \
    ASYNC_B128((g), (l), (base) + 512);                                         \
    ASYNC_B128((g), (l), (base) + 1024);                                        \
    ASYNC_B128((g), (l), (base) + 1536)
#define ISSUE16(g, l)                                                           \
    ISSUE4((g), (l), 0);                                                        \
    ISSUE4((g), (l), 2048);                                                     \
    ISSUE4((g), (l), 4096);                                                     \
    ISSUE4((g), (l), 6144)

__launch_bounds__(128, 1)
__global__ void paged_attn_wmma(const float* __restrict__ q,
                                const float* __restrict__ kc,
                                const float* __restrict__ vc,
                                const float* __restrict__ scale_p,
                                const int*   __restrict__ btab,
                                const int*   __restrict__ slens,
                                const float* __restrict__ slopes,
                                float*       __restrict__ out)
{
    const int kvh  = blockIdx.x;
    const int seq  = blockIdx.y;
    const int tid  = threadIdx.x;
    const int wave = tid >> 5;
    const int lane = tid & 31;
    const int hi   = lane >> 4;   // lane half (0/1)
    const int il   = lane & 15;   // lane-in-half

    __shared__ __align__(16) float q_s[BS * HD];          // 4 real head rows + 12 zero
    __shared__ __align__(16) float k_s[4][2][BS * HD];    // per-wave ping-pong K tiles
    __shared__ __align__(16) float v_s[4][2][BS * HD];    // per-wave ping-pong V tiles
    __shared__ __align__(16) float p_s[4][BS * 16];       // per-wave prob tile
    __shared__ float red_m[16];
    __shared__ float red_l[16];
    __shared__ __align__(16) float red_acc[16 * HD];

    const float scale  = scale_p[0];
    const int   seqlen = slens[seq];
    const int*  bt     = btab + seq * MAXB;

    // ---- stage Q (scaled) into LDS rows 0..3, zero rows 4..15 ----
    {
        const float4 qv = reinterpret_cast<const float4*>(
            q + (size_t)(seq * NH + kvh * GQA) * HD)[tid];
        float4 sq;
        sq.x = qv.x * scale; sq.y = qv.y * scale;
        sq.z = qv.z * scale; sq.w = qv.w * scale;
        reinterpret_cast<float4*>(q_s)[tid] = sq;
        #pragma unroll
        for (int r = 4; r < 16; ++r) q_s[r * HD + tid] = 0.0f;
    }
    __syncthreads();

    // ---- Q B-fragments (B[k][n] = Qs[head=n][4c+k], k = 2*hi + vgpr) ----
    v2f qb[HD / 4];
    #pragma unroll
    for (int c = 0; c < HD / 4; ++c) {
        qb[c].x = q_s[il * HD + 4 * c + 2 * hi];
        qb[c].y = q_s[il * HD + 4 * c + 2 * hi + 1];
    }

    const float slope = slopes[kvh * GQA + (il < GQA ? il : GQA - 1)];

    v8f acc[8];
    #pragma unroll
    for (int dc = 0; dc < 8; ++dc)
        #pragma unroll
        for (int j = 0; j < 8; ++j) acc[dc][j] = 0.0f;

    float m_run = -1e30f, l_run = 0.0f;
    float* pp = p_s[wave];
    const int ntiles = (seqlen + BS - 1) / BS;

    // ---- prologue: async-issue K+V for this wave's first tile (cnt -> 32) ----
    int cur = 0;
    if (wave < ntiles) {
        const int blk0 = bt[wave];
        const float* kg0 = kc + ((size_t)blk0 * NKVH + kvh) * (BS * HD);
        const float* vg0 = vc + ((size_t)blk0 * NKVH + kvh) * (BS * HD);
        ISSUE16(kg0 + lane * 4, &k_s[wave][0][lane * 4]);
        ISSUE16(vg0 + lane * 4, &v_s[wave][0][lane * 4]);
    }

    for (int t = wave; t < ntiles; t += 4) {
        const float* kb = k_s[wave][cur];
        const float* vb = v_s[wave][cur];

        // K tile landed; V tile (16 ops) may still be in flight.
        WAIT_ASYNC(16);

        // ---- S^T[token][head] = K(16x128) . Q^T(128x16), 32 WMMA k-chunks ----
        v8f c;
        #pragma unroll
        for (int j = 0; j < 8; ++j) c[j] = 0.0f;
        #pragma unroll
        for (int ch = 0; ch < HD / 4; ++ch) {
            v2f a;   // A[m=token][k] = K[token][4ch + 2*hi + vgpr]
            a.x = kb[il * HD + 4 * ch + 2 * hi];
            a.y = kb[il * HD + 4 * ch + 2 * hi + 1];
            c = __builtin_amdgcn_wmma_f32_16x16x4_f32(
                    false, a, false, qb[ch], (short)0, c, false, false);
        }

        // ---- ALiBi + mask + online softmax (lane owns head = il) ----
        float pv[8];
        float m_t = -1e30f;
        #pragma unroll
        for (int r = 0; r < 8; ++r) {
            const int token = r + 8 * hi;
            const int pos   = t * BS + token;
            float s = c[r] + slope * (float)(pos - (seqlen - 1));
            s = (pos < seqlen) ? s : -1e30f;
            pv[r] = s;
            m_t = fmaxf(m_t, s);
        }
        m_t = fmaxf(m_t, __shfl_xor(m_t, 16, 32));
        const float m_new = fmaxf(m_run, m_t);
        const float corr  = __expf(m_run - m_new);
        float s_t = 0.0f;
        #pragma unroll
        for (int r = 0; r < 8; ++r) { pv[r] = __expf(pv[r] - m_new); s_t += pv[r]; }
        s_t  += __shfl_xor(s_t, 16, 32);
        l_run = l_run * corr + s_t;
        m_run = m_new;
        #pragma unroll
        for (int dc = 0; dc < 8; ++dc)
            #pragma unroll
            for (int j = 0; j < 8; ++j) acc[dc][j] *= corr;

        // ---- P tile -> LDS (token-major, 16 heads incl. padded) ----
        #pragma unroll
        for (int r = 0; r < 8; ++r) pp[(r + 8 * hi) * 16 + il] = pv[r];

        // V tile landed.
        WAIT_ASYNC(0);

        // ---- prefetch next tile's K+V into the other buffer (overlaps PV) ----
        if (t + 4 < ntiles) {
            const int blkn = bt[t + 4];
            const float* kgn = kc + ((size_t)blkn * NKVH + kvh) * (BS * HD);
            const float* vgn = vc + ((size_t)blkn * NKVH + kvh) * (BS * HD);
            ISSUE16(kgn + lane * 4, &k_s[wave][cur ^ 1][lane * 4]);
            ISSUE16(vgn + lane * 4, &v_s[wave][cur ^ 1][lane * 4]);
        }

        // ---- O^T[d][head] += V^T . P : 8 d-chunks x 4 k-groups ----
        #pragma unroll
        for (int dc = 0; dc < 8; ++dc) {
            #pragma unroll
            for (int kg4 = 0; kg4 < 4; ++kg4) {
                v2f a, b;
                a.x = vb[(4 * kg4 + 2 * hi)     * HD + 16 * dc + il];
                a.y = vb[(4 * kg4 + 2 * hi + 1) * HD + 16 * dc + il];
                b.x = pp[(4 * kg4 + 2 * hi)     * 16 + il];
                b.y = pp[(4 * kg4 + 2 * hi + 1) * 16 + il];
                acc[dc] = __builtin_amdgcn_wmma_f32_16x16x4_f32(
                              false, a, false, b, (short)0, acc[dc], false, false);
            }
        }
        cur ^= 1;
    }

    // ---- publish per-wave partials (only real heads il<4) ----
    if (il < GQA && hi == 0) {
        red_m[wave * GQA + il] = m_run;
        red_l[wave * GQA + il] = l_run;
    }
    if (il < GQA) {
        #pragma unroll
        for (int dc = 0; dc < 8; ++dc)
            #pragma unroll
            for (int r = 0; r < 8; ++r) {
                const int d = 16 * dc + r + 8 * hi;
                red_acc[(wave * GQA + il) * HD + d] = acc[dc][r];
            }
    }
    __syncthreads();

    // ---- cross-wave log-sum-exp merge + write 4 heads x 128 dims ----
    #pragma unroll
    for (int i = 0; i < 4; ++i) {
        const int idx = tid + i * 128;   // 0..511
        const int h   = idx >> 7;
        const int d   = idx & 127;
        float M = -1e30f;
        #pragma unroll
        for (int w = 0; w < 4; ++w) M = fmaxf(M, red_m[w * GQA + h]);
        float Ls = 0.0f, o = 0.0f;
        #pragma unroll
        for (int w = 0; w < 4; ++w) {
            const float e = __expf(red_m[w * GQA + h] - M);
            Ls += e * red_l[w * GQA + h];
            o  += e * red_acc[(w * GQA + h) * HD + d];
        }
        out[(size_t)(seq * NH + kvh * GQA + h) * HD + d] = o / Ls;
    }
}

extern "C" void kernel_launch(void* const* d_in, const int* in_sizes, int n_in,
                              void* d_out, int out_size, void* d_ws, size_t ws_size,
                              hipStream_t stream) {
    (void)in_sizes; (void)n_in; (void)d_ws; (void)ws_size; (void)out_size;
    const float* q      = (const float*)d_in[0];
    const float* kc     = (const float*)d_in[1];
    const float* vc     = (const float*)d_in[2];
    const float* scale  = (const float*)d_in[4];
    const int*   btab   = (const int*)d_in[5];
    const int*   slens  = (const int*)d_in[6];
    const float* slopes = (const float*)d_in[9];
    float* out = (float*)d_out;

    dim3 grid(NKVH, NSEQ);
    paged_attn_wmma<<<grid, 128, 0, stream>>>(q, kc, vc, scale, btab, slens, slopes, out);
}